// MyCLUB_66649302499430
// MI455X (gfx1250) — compile-verified
//
#include <hip/hip_runtime.h>
#include <math.h>

typedef __attribute__((ext_vector_type(2))) float v2f;
typedef __attribute__((ext_vector_type(8))) float v8f;

#define N_SAMP 1024
#define DIM 256
#define LDS_STRIDE 260   // 256 + 4: makes 16 LDS rows hit 16 distinct bank groups
#define NBLK (N_SAMP / 16)

// ---------------------------------------------------------------------------
// Kernel A: per-column statistics of y and z.
//   c0[d] = mean_j y[j,d]^2 + mean_j z[j,d]^2
//   c1[d] = mean_j y[j,d]   + mean_j z[j,d]
// 256 threads, one column each; loads are coalesced across the wave.
// ---------------------------------------------------------------------------
__global__ __launch_bounds__(256) void club_colstats(const float* __restrict__ y,
                                                     const float* __restrict__ z,
                                                     float* __restrict__ c0,
                                                     float* __restrict__ c1) {
  const int d = threadIdx.x;
  float sy = 0.f, sz = 0.f, syy = 0.f, szz = 0.f;
  for (int j = 0; j < N_SAMP; ++j) {
    float yv = y[j * DIM + d];
    float zv = z[j * DIM + d];
    sy += yv; sz += zv;
    syy += yv * yv; szz += zv * zv;
  }
  const float inv = 1.0f / (float)N_SAMP;
  c0[d] = (syy + szz) * inv;
  c1[d] = (sy + sz) * inv;
}

// ---------------------------------------------------------------------------
// Kernel B: one wave32 per 16-row stripe of the batch.
//   layer1 (mu & lv): H = relu(x @ W1^T + b1) via v_wmma_f32_16x16x4_f32 -> LDS
//   layer2: mu / logvar tiles, fused tanh/exp + factored CLUB terms,
//   row-reduction via xor shuffles, per-block partial sum out.
// ---------------------------------------------------------------------------
__global__ __launch_bounds__(32) void club_main(
    const float* __restrict__ x,  const float* __restrict__ y,  const float* __restrict__ z,
    const float* __restrict__ w1mu, const float* __restrict__ b1mu,
    const float* __restrict__ w2mu, const float* __restrict__ b2mu,
    const float* __restrict__ w1lv, const float* __restrict__ b1lv,
    const float* __restrict__ w2lv, const float* __restrict__ b2lv,
    const float* __restrict__ c0v,  const float* __restrict__ c1v,
    float* __restrict__ blockSums) {
  __shared__ float Hmu[16 * LDS_STRIDE];
  __shared__ float Hlv[16 * LDS_STRIDE];

  const int lane = threadIdx.x;   // 0..31
  const int g    = lane >> 4;     // half-wave: selects K sub-pair / M upper half
  const int lr   = lane & 15;     // row (A) / col (B,D) within tile
  const int i0   = blockIdx.x * 16;

  // ---------------- layer 1: both branches ----------------
  for (int m = 0; m < 2; ++m) {
    const float* W1 = m ? w1lv : w1mu;
    const float* B1 = m ? b1lv : b1mu;
    float*       H  = m ? Hlv  : Hmu;
    // A fragment base: lane holds X[i0+lr, 4*kk + 2*g + {0,1}]
    const float* arow = x + (size_t)(i0 + lr) * DIM + 2 * g;
    for (int n = 0; n < 16; ++n) {
      v8f acc = {};
      // B fragment base: lane holds W1[n*16+lr, 4*kk + 2*g + {0,1}]
      const float* brow = W1 + (size_t)(n * 16 + lr) * DIM + 2 * g;
      #pragma unroll 4
      for (int kk = 0; kk < 64; ++kk) {
        v2f a = *(const v2f*)(arow + kk * 4);
        v2f b = *(const v2f*)(brow + kk * 4);
        acc = __builtin_amdgcn_wmma_f32_16x16x4_f32(false, a, false, b,
                                                    (short)0, acc, false, false);
      }
      const float bias = B1[n * 16 + lr];
      #pragma unroll
      for (int r = 0; r < 8; ++r) {
        float h = acc[r] + bias;                    // D: (M = r + 8g, Ncol = lr)
        H[(r + 8 * g) * LDS_STRIDE + n * 16 + lr] = h > 0.f ? h : 0.f;
      }
    }
  }
  __syncthreads();

  // ---------------- layer 2 + fused epilogue ----------------
  float rowsum[8];
  #pragma unroll
  for (int r = 0; r < 8; ++r) rowsum[r] = 0.f;

  const float* amu_base = &Hmu[lr * LDS_STRIDE + 2 * g];
  const float* alv_base = &Hlv[lr * LDS_STRIDE + 2 * g];

  for (int n = 0; n < 16; ++n) {
    v8f accmu = {}, acclv = {};
    const float* bmu_row = w2mu + (size_t)(n * 16 + lr) * DIM + 2 * g;
    const float* blv_row = w2lv + (size_t)(n * 16 + lr) * DIM + 2 * g;
    #pragma unroll 4
    for (int kk = 0; kk < 64; ++kk) {
      v2f a  = *(const v2f*)(amu_base + kk * 4);     // ds_load_b64 (bank-clean)
      v2f b  = *(const v2f*)(bmu_row  + kk * 4);
      accmu = __builtin_amdgcn_wmma_f32_16x16x4_f32(false, a,  false, b,
                                                    (short)0, accmu, false, false);
      v2f a2 = *(const v2f*)(alv_base + kk * 4);
      v2f b2 = *(const v2f*)(blv_row  + kk * 4);
      acclv = __builtin_amdgcn_wmma_f32_16x16x4_f32(false, a2, false, b2,
                                                    (short)0, acclv, false, false);
    }
    const int   col    = n * 16 + lr;
    const float biasmu = b2mu[col];
    const float biaslv = b2lv[col];
    const float c0     = c0v[col];
    const float c1     = c1v[col];
    #pragma unroll
    for (int r = 0; r < 8; ++r) {
      const int row = i0 + r + 8 * g;
      float mu = accmu[r] + biasmu;
      float lv = tanhf(acclv[r] + biaslv);
      float iv = 0.5f * expf(-lv);                   // 1 / (2 exp(logvar))
      float yv = y[(size_t)row * DIM + col];
      float zv = z[(size_t)row * DIM + col];
      float dy = mu - yv, dz = mu - zv;
      float negt = c0 - 2.f * mu * c1 + 2.f * mu * mu;   // factored mean_j sq
      float post = dy * dy + dz * dz;
      rowsum[r] += iv * (negt - post);               // positive - negative contrib
    }
  }

  // Reduce over the 16 lanes of each half-wave (same row, different cols).
  #pragma unroll
  for (int r = 0; r < 8; ++r) {
    float v = rowsum[r];
    v += __shfl_xor(v, 1, 32);
    v += __shfl_xor(v, 2, 32);
    v += __shfl_xor(v, 4, 32);
    v += __shfl_xor(v, 8, 32);
    rowsum[r] = v;                                   // full diff for row r + 8g
  }
  float s = 0.f;
  #pragma unroll
  for (int r = 0; r < 8; ++r) s += rowsum[r] > 0.f ? rowsum[r] : 0.f;  // relu
  s += __shfl_xor(s, 16, 32);                        // combine the two halves
  if (lane == 0) blockSums[blockIdx.x] = s;          // sum of relu over 16 rows
}

// ---------------------------------------------------------------------------
// Kernel C: ordered (deterministic) final reduction of 64 block partials.
// ---------------------------------------------------------------------------
__global__ void club_final(const float* __restrict__ blockSums, float* __restrict__ out) {
  if (threadIdx.x == 0) {
    float t = 0.f;
    for (int i = 0; i < NBLK; ++i) t += blockSums[i];
    out[0] = t / (float)N_SAMP;
  }
}

// ---------------------------------------------------------------------------
extern "C" void kernel_launch(void* const* d_in, const int* in_sizes, int n_in,
                              void* d_out, int out_size, void* d_ws, size_t ws_size,
                              hipStream_t stream) {
  const float* x    = (const float*)d_in[0];
  const float* y    = (const float*)d_in[1];
  const float* z    = (const float*)d_in[2];
  const float* w1mu = (const float*)d_in[3];
  const float* b1mu = (const float*)d_in[4];
  const float* w2mu = (const float*)d_in[5];
  const float* b2mu = (const float*)d_in[6];
  const float* w1lv = (const float*)d_in[7];
  const float* b1lv = (const float*)d_in[8];
  const float* w2lv = (const float*)d_in[9];
  const float* b2lv = (const float*)d_in[10];

  float* ws = (float*)d_ws;
  float* c0 = ws;              // 256 floats
  float* c1 = ws + DIM;        // 256 floats
  float* bs = ws + 2 * DIM;    // 64 floats

  club_colstats<<<1, DIM, 0, stream>>>(y, z, c0, c1);
  club_main<<<NBLK, 32, 0, stream>>>(x, y, z, w1mu, b1mu, w2mu, b2mu,
                                     w1lv, b1lv, w2lv, b2lv, c0, c1, bs);
  club_final<<<1, 32, 0, stream>>>(bs, (float*)d_out);
}